// Yolo_loss_81767587381937
// MI455X (gfx1250) — compile-verified
//
#include <hip/hip_runtime.h>

typedef __attribute__((ext_vector_type(2))) float v2f;
typedef __attribute__((ext_vector_type(8))) float v8f;

// ---------------------------------------------------------------------------
// Wave32 sum-reduction using the CDNA5 matrix pipe:
//   A (16x4): A[m][0] = p(lane m), A[m][1] = 0, A[m][2] = p(lane m+16), A[m][3] = 0
//   B (4x16): all ones  ->  D[m][n] = p(m) + p(m+16)
// D layout: lanes 0-15 hold rows 0..7 in v[0..7]; lanes 16-31 hold rows 8..15.
// Sum of own 8 D regs + shfl_xor(16) => full wave sum in every lane.
// Fixed-order hardware accumulation => deterministic.
// Requires EXEC == all ones (call only from convergent code).
// ---------------------------------------------------------------------------
__device__ __forceinline__ float wave_reduce_wmma(float v) {
  v2f a; a.x = v;    a.y = 0.0f;
  v2f b; b.x = 1.0f; b.y = 1.0f;
  v8f c = {};
  c = __builtin_amdgcn_wmma_f32_16x16x4_f32(
      /*neg_a=*/false, a, /*neg_b=*/false, b,
      /*c_mod=*/(short)0, c, /*reuse_a=*/false, /*reuse_b=*/false);
  float s = c[0] + c[1] + c[2] + c[3] + c[4] + c[5] + c[6] + c[7];
  s += __shfl_xor(s, 16, 32);
  return s;
}

__device__ __forceinline__ float iou_f(float acx, float acy, float aw, float ah,
                                       float bcx, float bcy, float bw, float bh) {
  float axm = acx - 0.5f * aw, axM = acx + 0.5f * aw;
  float aym = acy - 0.5f * ah, ayM = acy + 0.5f * ah;
  float bxm = bcx - 0.5f * bw, bxM = bcx + 0.5f * bw;
  float bym = bcy - 0.5f * bh, byM = bcy + 0.5f * bh;
  float iw = fmaxf(fminf(axM, bxM) - fmaxf(axm, bxm), 0.0f);
  float ih = fmaxf(fminf(ayM, byM) - fmaxf(aym, bym), 0.0f);
  float inter = iw * ih;
  float uni   = aw * ah + bw * bh - inter + 1e-10f;
  return inter / uni;
}

// One cell per thread (grid-stride). 30 contiguous f32 per tensor per cell
// (120 B, 8-byte aligned) -> 15 global_load_b64 each; dense across lanes.
__global__ __launch_bounds__(256) void yolo_partial_kernel(
    const float* __restrict__ preds, const float* __restrict__ labels,
    float* __restrict__ partials, long long ncells) {
  const long long stride = (long long)gridDim.x * blockDim.x;
  float acc = 0.0f;

  for (long long cell = (long long)blockIdx.x * blockDim.x + threadIdx.x;
       cell < ncells; cell += stride) {
    const size_t base = (size_t)cell * 30;
    const v2f* p2 = (const v2f*)(preds + base);
    const v2f* l2 = (const v2f*)(labels + base);
    float P[30], L[30];
#pragma unroll
    for (int k = 0; k < 15; ++k) {
      v2f t = p2[k]; P[2 * k] = t.x; P[2 * k + 1] = t.y;
      v2f u = l2[k]; L[2 * k] = u.x; L[2 * k + 1] = u.y;
    }

    // faithful quirk: label bbox x-coord read transposed within the 7x7 tile
    long long b = cell / 49;
    int rem = (int)(cell - b * 49);
    int y = rem / 7, x = rem - y * 7;
    float lb0 = labels[(size_t)(b * 49 + (long long)x * 7 + y) * 30];

    float iou1 = iou_f(P[0], P[1], P[2], P[3], lb0, L[1], L[2], L[3]);
    float iou2 = iou_f(P[5], P[6], P[7], P[8], lb0, L[1], L[2], L[3]);

    float d0 = P[0] - L[0], d1 = P[1] - L[1];
    float s2 = sqrtf(P[2]) - sqrtf(L[2]), s3 = sqrtf(P[3]) - sqrtf(L[3]);
    float c1 = iou1 - P[4];
    float b1v = 5.0f * (d0 * d0 + d1 * d1) + s2 * s2 + s3 * s3 + c1 * c1 +
                0.5f * P[9] * P[9];

    float e5 = P[5] - L[5], e6 = P[6] - L[6];
    float s7 = sqrtf(P[7]) - sqrtf(L[7]), s8 = sqrtf(P[8]) - sqrtf(L[8]);
    float c2 = iou2 - P[9];
    float b2v = 5.0f * (e5 * e5 + e6 * e6) + s7 * s7 + s8 * s8 + c2 * c2 +
                0.5f * P[4] * P[4];

    float cls = 0.0f;
#pragma unroll
    for (int c = 10; c < 30; ++c) {
      float d = L[c] - P[c];
      cls += d * d;
    }

    float obj_loss   = ((iou1 > iou2) ? b1v : b2v) + cls;
    float noobj_loss = 0.5f * (P[4] * P[4] + P[9] * P[9]);
    acc += (L[4] == 1.0f) ? obj_loss : noobj_loss;
  }

  // convergent here: reduce wave via WMMA, then block via LDS (fixed order)
  float wsum = wave_reduce_wmma(acc);
  __shared__ float smem[8];
  const int lane = threadIdx.x & 31;
  const int wid  = threadIdx.x >> 5;
  if (lane == 0) smem[wid] = wsum;
  __syncthreads();
  if (threadIdx.x == 0) {
    float t = 0.0f;
    const int nw = blockDim.x >> 5;
    for (int i = 0; i < nw; ++i) t += smem[i];
    partials[blockIdx.x] = t;
  }
}

// Single block: deterministically fold the per-block partials, scale by 1/B.
__global__ __launch_bounds__(256) void yolo_final_kernel(
    const float* __restrict__ partials, int n, float* __restrict__ out,
    float inv_batch) {
  float acc = 0.0f;
  for (int i = threadIdx.x; i < n; i += blockDim.x) acc += partials[i];
  float wsum = wave_reduce_wmma(acc);
  __shared__ float smem[8];
  const int lane = threadIdx.x & 31;
  const int wid  = threadIdx.x >> 5;
  if (lane == 0) smem[wid] = wsum;
  __syncthreads();
  if (threadIdx.x == 0) {
    float t = 0.0f;
    const int nw = blockDim.x >> 5;
    for (int i = 0; i < nw; ++i) t += smem[i];
    out[0] = t * inv_batch;  // overwrite: no zero-init / atomics needed
  }
}

extern "C" void kernel_launch(void* const* d_in, const int* in_sizes, int n_in,
                              void* d_out, int out_size, void* d_ws,
                              size_t ws_size, hipStream_t stream) {
  const float* preds  = (const float*)d_in[0];
  const float* labels = (const float*)d_in[1];
  float* out = (float*)d_out;

  const long long nelem  = (long long)in_sizes[0];
  const long long ncells = nelem / 30;          // B * 7 * 7
  const long long batch  = nelem / (7 * 7 * 30);

  const int block = 256;
  long long want = (ncells + block - 1) / block;
  long long cap  = (long long)(ws_size / sizeof(float));
  if (cap > 4096) cap = 4096;                   // grid-stride covers the rest
  int grid = (int)((want < cap) ? want : cap);
  if (grid < 1) grid = 1;

  float* partials = (float*)d_ws;

  yolo_partial_kernel<<<grid, block, 0, stream>>>(preds, labels, partials,
                                                  ncells);
  yolo_final_kernel<<<1, block, 0, stream>>>(
      partials, grid, out, 1.0f / (float)batch);
}